// Rasterize_31318901522754
// MI455X (gfx1250) — compile-verified
//
#include <hip/hip_runtime.h>

// ---------------- problem constants ----------------
#define NMESH   8
#define NVERT   10000
#define NFACE   20000
#define GD      128
#define VOXELS  (GD*GD*GD)          // 2097152 per mesh
#define NSAMP   36                  // barycentric samples per triangle (S=8)
#define FTILE   16                  // faces per WMMA tile (M dim)
#define NFT     (NFACE/FTILE)       // 1250 face tiles per mesh (exact)
#define NPT     3                   // sample tiles per face tile (36 -> 16,16,4)
#define TASKS   (NMESH*NFT*NPT)     // 30000 wave-tasks

typedef __attribute__((ext_vector_type(2))) float v2f;
typedef __attribute__((ext_vector_type(8))) float v8f;

// BARY grid for S=8: lin(i)=(2i+1)/16; pairs (a_i,b_j) with i+j<=7, row-major
// in i, padded with zeros to 48 entries so lane indexing never goes OOB.
__constant__ float BARY_A[48] = {
  0.0625f,0.0625f,0.0625f,0.0625f,0.0625f,0.0625f,0.0625f,0.0625f,
  0.1875f,0.1875f,0.1875f,0.1875f,0.1875f,0.1875f,0.1875f,
  0.3125f,0.3125f,0.3125f,0.3125f,0.3125f,0.3125f,
  0.4375f,0.4375f,0.4375f,0.4375f,0.4375f,
  0.5625f,0.5625f,0.5625f,0.5625f,
  0.6875f,0.6875f,0.6875f,
  0.8125f,0.8125f,
  0.9375f,
  0,0,0,0,0,0,0,0,0,0,0,0
};
__constant__ float BARY_B[48] = {
  0.0625f,0.1875f,0.3125f,0.4375f,0.5625f,0.6875f,0.8125f,0.9375f,
  0.0625f,0.1875f,0.3125f,0.4375f,0.5625f,0.6875f,0.8125f,
  0.0625f,0.1875f,0.3125f,0.4375f,0.5625f,0.6875f,
  0.0625f,0.1875f,0.3125f,0.4375f,0.5625f,
  0.0625f,0.1875f,0.3125f,0.4375f,
  0.0625f,0.1875f,0.3125f,
  0.0625f,0.1875f,
  0.0625f,
  0,0,0,0,0,0,0,0,0,0,0,0
};

// ---------------- zero / clamp passes ----------------
__global__ void zero_kernel(float4* __restrict__ out) {
  int i = blockIdx.x * blockDim.x + threadIdx.x;
  out[i] = make_float4(0.f, 0.f, 0.f, 0.f);
}

__global__ void clamp_kernel(float4* __restrict__ out) {
  int i = blockIdx.x * blockDim.x + threadIdx.x;
  float4 v = out[i];
  v.x = fminf(fmaxf(v.x, 0.f), 1.f);
  v.y = fminf(fmaxf(v.y, 0.f), 1.f);
  v.z = fminf(fmaxf(v.z, 0.f), 1.f);
  v.w = fminf(fmaxf(v.w, 0.f), 1.f);
  out[i] = v;
}

// ---------------- WMMA sample + trilinear splat ----------------
// One wave per (mesh, 16-face tile, 16-sample tile).
//   A (16x4 f32)  : M=face, K=corner. lanes 0-15 hold K=0,1; lanes 16-31 hold K=2,(0)
//   B (4x16 f32)  : K=corner, N=sample. lanes 0-15 hold K=0,1 rows; lanes 16-31 K=2,(0)
//   D (16x16 f32) : lane&15 = sample N, VGPR m = face row (lanes>=16 -> M+8)
__global__ __launch_bounds__(256) void splat_kernel(const float* __restrict__ verts,
                                                    const int*   __restrict__ faces,
                                                    float*       __restrict__ vol) {
  const int lane = threadIdx.x & 31;
  const int task = blockIdx.x * 8 + (threadIdx.x >> 5);   // 8 waves / block
  if (task >= TASKS) return;                              // wave-uniform

  const int mesh  = task / (NFT * NPT);
  int r           = task - mesh * (NFT * NPT);
  const int ftile = r / NPT;
  const int ptile = r - ftile * NPT;

  const int half = lane >> 4;       // 0: lower K half, 1: upper K half
  const int l15  = lane & 15;

  // ---- A matrices (one per output coordinate) ----
  const int f  = ftile * FTILE + l15;
  const int* fp = faces + ((size_t)mesh * NFACE + f) * 3;
  const int i0 = fp[0], i1 = fp[1], i2 = fp[2];
  const int j0 = half ? i2 : i0;                 // K slot 0/2
  const int j1 = half ? i0 : i1;                 // K slot 1 (upper half masked to 0)
  const float msk = half ? 0.0f : 1.0f;
  const float* vb = verts + (size_t)mesh * NVERT * 3;
  const float p0x = vb[j0*3+0], p0y = vb[j0*3+1], p0z = vb[j0*3+2];
  const float p1x = vb[j1*3+0]*msk, p1y = vb[j1*3+1]*msk, p1z = vb[j1*3+2]*msk;
  v2f aX; aX.x = p0x; aX.y = p1x;
  v2f aY; aY.x = p0y; aY.y = p1y;
  v2f aZ; aZ.x = p0z; aZ.y = p1z;

  // ---- B matrix: BARY^T tile ----
  const int s  = ptile * 16 + l15;               // sample index (may be pad)
  const float ba = BARY_A[s];
  const float bb = BARY_B[s];
  const float bc = (s < NSAMP) ? (1.0f - ba - bb) : 0.0f;
  v2f b;
  b.x = half ? bc   : ba;
  b.y = half ? 0.0f : bb;

  v8f c = {0.f,0.f,0.f,0.f,0.f,0.f,0.f,0.f};
  // D = A x B : 16 faces x 16 samples of one coordinate each
  v8f dX = __builtin_amdgcn_wmma_f32_16x16x4_f32(false, aX, false, b, (short)0, c, false, false);
  v8f dY = __builtin_amdgcn_wmma_f32_16x16x4_f32(false, aY, false, b, (short)0, c, false, false);
  v8f dZ = __builtin_amdgcn_wmma_f32_16x16x4_f32(false, aZ, false, b, (short)0, c, false, false);

  if (s >= NSAMP) return;                        // pad samples: no contribution
  float* v = vol + (size_t)mesh * VOXELS;

  // Each lane splats 8 (face,sample) points: D rows m=0..7 (lanes>=16: faces +8)
  #pragma unroll
  for (int m = 0; m < 8; ++m) {
    const float xd = 63.5f * (dX[m] + 1.0f);     // (128-1)*(p+1)*0.5
    const float xh = 63.5f * (dY[m] + 1.0f);
    const float xw = 63.5f * (dZ[m] + 1.0f);
    const float fd = floorf(xd), fh = floorf(xh), fw = floorf(xw);
    const float rd = xd - fd, rh = xh - fh, rw = xw - fw;
    const int id = (int)fd, ih = (int)fh, iw = (int)fw;
    #pragma unroll
    for (int dz = 0; dz < 2; ++dz) {
      const int   zi = min(max(id + dz, 0), GD - 1);
      const float wz = dz ? rd : 1.0f - rd;
      #pragma unroll
      for (int dy = 0; dy < 2; ++dy) {
        const int   yi = min(max(ih + dy, 0), GD - 1);
        const float wy = dy ? rh : 1.0f - rh;
        #pragma unroll
        for (int dx = 0; dx < 2; ++dx) {
          const int   xi = min(max(iw + dx, 0), GD - 1);
          const float wx = dx ? rw : 1.0f - rw;
          atomicAdd(v + ((((zi << 7) + yi) << 7) + xi), wz * wy * wx);
        }
      }
    }
  }
}

// ---------------- launch ----------------
extern "C" void kernel_launch(void* const* d_in, const int* in_sizes, int n_in,
                              void* d_out, int out_size, void* d_ws, size_t ws_size,
                              hipStream_t stream) {
  const float* verts = (const float*)d_in[0];   // [8,10000,3] f32
  const int*   faces = (const int*)d_in[1];     // [8,20000,3] i32
  float*       out   = (float*)d_out;           // [8,128,128,128] f32

  const int n4 = (NMESH * VOXELS) / 4;          // 4,194,304 float4s (exact)
  zero_kernel <<<n4 / 256, 256, 0, stream>>>((float4*)out);
  splat_kernel<<<(TASKS + 7) / 8, 256, 0, stream>>>(verts, faces, out);
  clamp_kernel<<<n4 / 256, 256, 0, stream>>>((float4*)out);
}